// BahdanauAttention_64390149701814
// MI455X (gfx1250) — compile-verified
//
#include <hip/hip_runtime.h>
#include <hip/hip_bf16.h>
#include <math.h>

// Problem constants (from reference): B=32, T=4096, H=512, QS=512
constexpr int Bn = 32;
constexpr int Tn = 4096;
constexpr int Hn = 512;
constexpr int QSn = 512;
constexpr int TCHUNK = 32;           // t-chunks for context partials
constexpr int TROWS = Tn / TCHUNK;   // 128 rows per chunk

typedef __attribute__((ext_vector_type(2))) float v2f;
typedef __attribute__((ext_vector_type(8))) float v8f;

// gfx1250 has a hardware transcendental v_tanh_f32; use it so the 64M-element
// tanh stream stays ~1 VALU op/elem and phase B remains memory-bound.
static __device__ __forceinline__ float fast_tanh(float x) {
#if __has_builtin(__builtin_amdgcn_tanhf)
    return __builtin_amdgcn_tanhf(x);
#elif __has_builtin(__builtin_amdgcn_tanh_f32)
    return __builtin_amdgcn_tanh_f32(x);
#else
    return tanhf(x);
#endif
}

// ---------------------------------------------------------------------------
// Kernel 1: q[b,h] = sum_s query[b,s] * Wq[h,s]   (32x512 GEMM, K=512)
// Uses V_WMMA_F32_16X16X4_F32. One wave per 16x16 output tile:
//   2 M-tiles (B=32) x 32 N-tiles (H=512) = 64 waves, 128 WMMA ops per wave.
// A (16x4 f32): lane l holds M = l%16, K = 2*(l/16) + j   (j = vgpr 0/1)
// B (4x16 f32): lane l holds N = l%16, K = 2*(l/16) + j   (mirror of A)
// C (16x16 f32): lane l, vgpr r holds M = r + 8*(l/16), N = l%16
// ---------------------------------------------------------------------------
__global__ void __launch_bounds__(128)
qproj_wmma_kernel(const float* __restrict__ query,  // [B, QS]
                  const float* __restrict__ Wq,     // [H, QS]
                  float* __restrict__ qout)         // [B, H]
{
    const int lane  = threadIdx.x & 31;
    const int wave  = blockIdx.x * (blockDim.x >> 5) + (threadIdx.x >> 5); // 0..63
    const int tileM = wave >> 5;   // 0..1
    const int tileN = wave & 31;   // 0..31
    const int lm    = lane & 15;
    const int hi    = lane >> 4;

    const float* arow = query + (size_t)(tileM * 16 + lm) * QSn;
    const float* brow = Wq    + (size_t)(tileN * 16 + lm) * QSn;

    v8f c = {};
    for (int k0 = 0; k0 < QSn; k0 += 4) {
        v2f a, b;
        a.x = arow[k0 + 2 * hi];
        a.y = arow[k0 + 2 * hi + 1];
        b.x = brow[k0 + 2 * hi];
        b.y = brow[k0 + 2 * hi + 1];
        c = __builtin_amdgcn_wmma_f32_16x16x4_f32(
                /*neg_a=*/false, a, /*neg_b=*/false, b,
                /*c_mod=*/(short)0, c, /*reuse_a=*/false, /*reuse_b=*/false);
    }

    const int h = tileN * 16 + lm;
#pragma unroll
    for (int r = 0; r < 8; ++r) {
        const int bidx = tileM * 16 + hi * 8 + r;
        qout[(size_t)bidx * Hn + h] = c[r];
    }
}

// ---------------------------------------------------------------------------
// Kernel 2: scores[b,t] = sum_h tanh(q[b,h] + key[b,t,h]) * We[h]
// One wave per row of key (coalesced float4 loads, 512B/wave/iter), 16
// consecutive rows per wave (same b -> q row cached in registers).
// Streams 256 MB -> this is the bandwidth-critical kernel (with kernel 4).
// ---------------------------------------------------------------------------
__global__ void __launch_bounds__(256)
scores_kernel(const float* __restrict__ q,      // [B, H]
              const float* __restrict__ key,    // [B, T, H]
              const float* __restrict__ We,     // [H]
              float* __restrict__ scores)       // [B, T]
{
    __shared__ float sWe[Hn];
    for (int i = threadIdx.x; i < Hn; i += blockDim.x) sWe[i] = We[i];
    __syncthreads();

    const int lane  = threadIdx.x & 31;
    const int gwave = (blockIdx.x * blockDim.x + threadIdx.x) >> 5;  // 0..8191
    const int row0  = gwave * 16;                // 16 rows per wave, same b
    const int b     = row0 >> 12;                // T = 4096

    float4 qv[4], wv[4];
#pragma unroll
    for (int i = 0; i < 4; ++i) {
        const int h = i * 128 + lane * 4;
        qv[i] = *(const float4*)(q + (size_t)b * Hn + h);
        wv[i] = *(const float4*)(&sWe[h]);
    }

    for (int r = 0; r < 16; ++r) {
        const float* krow = key + (size_t)(row0 + r) * Hn;
        // speculative prefetch of the next row (global_prefetch_b8)
        __builtin_prefetch(krow + Hn + lane * 4, 0, 0);
        float acc = 0.f;
#pragma unroll
        for (int i = 0; i < 4; ++i) {
            const int h = i * 128 + lane * 4;
            const float4 kv = *(const float4*)(krow + h);
            acc += fast_tanh(qv[i].x + kv.x) * wv[i].x;
            acc += fast_tanh(qv[i].y + kv.y) * wv[i].y;
            acc += fast_tanh(qv[i].z + kv.z) * wv[i].z;
            acc += fast_tanh(qv[i].w + kv.w) * wv[i].w;
        }
#pragma unroll
        for (int off = 16; off > 0; off >>= 1)
            acc += __shfl_xor(acc, off, 32);
        if (lane == 0) scores[row0 + r] = acc;
    }
}

// ---------------------------------------------------------------------------
// Kernel 3: in-place numerically-stable softmax over T per batch.
// One block (256 threads) per b; 16 elements per thread held in registers.
// ---------------------------------------------------------------------------
__global__ void __launch_bounds__(256)
softmax_kernel(float* __restrict__ sa)   // [B, T] scores in, alphas out
{
    float* row = sa + (size_t)blockIdx.x * Tn;
    const int tid  = threadIdx.x;
    const int lane = tid & 31;
    const int wid  = tid >> 5;
    __shared__ float red[8];

    float vals[16];
    float lmax = -INFINITY;
#pragma unroll
    for (int i = 0; i < 16; ++i) {
        vals[i] = row[tid + i * 256];
        lmax = fmaxf(lmax, vals[i]);
    }
#pragma unroll
    for (int off = 16; off > 0; off >>= 1)
        lmax = fmaxf(lmax, __shfl_xor(lmax, off, 32));
    if (lane == 0) red[wid] = lmax;
    __syncthreads();
    float gmax = red[0];
#pragma unroll
    for (int w = 1; w < 8; ++w) gmax = fmaxf(gmax, red[w]);
    __syncthreads();

    float lsum = 0.f;
#pragma unroll
    for (int i = 0; i < 16; ++i) {
        const float e = __expf(vals[i] - gmax);
        vals[i] = e;
        lsum += e;
    }
#pragma unroll
    for (int off = 16; off > 0; off >>= 1)
        lsum += __shfl_xor(lsum, off, 32);
    if (lane == 0) red[wid] = lsum;
    __syncthreads();
    float gsum = 0.f;
#pragma unroll
    for (int w = 0; w < 8; ++w) gsum += red[w];
    const float inv = 1.0f / gsum;

#pragma unroll
    for (int i = 0; i < 16; ++i)
        row[tid + i * 256] = vals[i] * inv;
}

// ---------------------------------------------------------------------------
// Kernel 4: partial context. Block (b, c) streams value rows t in
// [c*128, c*128+128) with fully coalesced loads; alphas staged in LDS.
// 1024 blocks in flight -> enough parallelism for 23.3 TB/s.
// ---------------------------------------------------------------------------
__global__ void __launch_bounds__(256)
context_partial_kernel(const float* __restrict__ alphas,  // [B, T]
                       const float* __restrict__ value,   // [B, T, H]
                       float* __restrict__ part)          // [TCHUNK, B, H]
{
    const int b = blockIdx.x;   // 0..31
    const int c = blockIdx.y;   // 0..31
    const int t0 = c * TROWS;

    __shared__ float sA[TROWS];
    if (threadIdx.x < TROWS)
        sA[threadIdx.x] = alphas[(size_t)b * Tn + t0 + threadIdx.x];
    __syncthreads();

    const float* vbase = value + ((size_t)b * Tn + t0) * Hn;
    float acc0 = 0.f, acc1 = 0.f;
    for (int i = 0; i < TROWS; ++i) {
        const float a = sA[i];
        const float* vr = vbase + (size_t)i * Hn;
        __builtin_prefetch(vr + Hn + threadIdx.x, 0, 0);
        acc0 += a * vr[threadIdx.x];
        acc1 += a * vr[threadIdx.x + 256];
    }
    float* p = part + ((size_t)c * Bn + b) * Hn;
    p[threadIdx.x]       = acc0;
    p[threadIdx.x + 256] = acc1;
}

// ---------------------------------------------------------------------------
// Kernel 5: deterministic fixed-order reduction of chunk partials.
// ---------------------------------------------------------------------------
__global__ void __launch_bounds__(256)
context_reduce_kernel(const float* __restrict__ part,  // [TCHUNK, B, H]
                      float* __restrict__ ctx)         // [B, H]
{
    const int b = blockIdx.x;
    const int d = threadIdx.x;
    float s0 = 0.f, s1 = 0.f;
    for (int c = 0; c < TCHUNK; ++c) {
        const float* p = part + ((size_t)c * Bn + b) * Hn;
        s0 += p[d];
        s1 += p[d + 256];
    }
    ctx[(size_t)b * Hn + d]       = s0;
    ctx[(size_t)b * Hn + d + 256] = s1;
}

// ---------------------------------------------------------------------------
// Host launcher
// ---------------------------------------------------------------------------
extern "C" void kernel_launch(void* const* d_in, const int* in_sizes, int n_in,
                              void* d_out, int out_size, void* d_ws, size_t ws_size,
                              hipStream_t stream) {
    (void)in_sizes; (void)n_in; (void)out_size; (void)ws_size;

    // Inputs in setup_inputs() order:
    const float* query = (const float*)d_in[0];   // [B,1,QS]
    const float* key   = (const float*)d_in[1];   // [B,T,H]
    const float* value = (const float*)d_in[2];   // [B,T,H]
    // d_in[3] = mask (unused by reference forward)
    const float* Wq    = (const float*)d_in[4];   // [H,QS]
    const float* We    = (const float*)d_in[5];   // [H]

    // Outputs: context [B,1,H] then alphas [B,T], concatenated flat.
    float* ctx    = (float*)d_out;                 // B*H floats
    float* alphas = (float*)d_out + Bn * Hn;       // B*T floats (scores in-place)

    // Workspace: q projection (64 KB) + context partials (2 MB).
    float* qws  = (float*)d_ws;                    // B*H
    float* part = qws + Bn * Hn;                   // TCHUNK*B*H

    // 1) q = query @ Wq.T  (WMMA f32 16x16x4; 64 waves)
    qproj_wmma_kernel<<<16, 128, 0, stream>>>(query, Wq, qws);

    // 2) raw scores into the alphas slot (streams 256 MB of key)
    scores_kernel<<<1024, 256, 0, stream>>>(qws, key, We, alphas);

    // 3) softmax in-place over T per batch
    softmax_kernel<<<Bn, 256, 0, stream>>>(alphas);

    // 4) partial contexts (streams 256 MB of value)
    context_partial_kernel<<<dim3(Bn, TCHUNK), 256, 0, stream>>>(alphas, value, part);

    // 5) deterministic reduce -> context
    context_reduce_kernel<<<Bn, 256, 0, stream>>>(part, ctx);
}